// DepthPriorLoss_26121991094299
// MI455X (gfx1250) — compile-verified
//
#include <hip/hip_runtime.h>

// CDNA5 / gfx1250, wave32. Uses V_WMMA_F32_16X16X4_F32 for the RANSAC
// residual grid (a K=3 GEMM padded to K=4).

typedef __attribute__((ext_vector_type(2))) float v2f;
typedef __attribute__((ext_vector_type(8))) float v8f;

// ---------------- workspace layout (32-bit words) ----------------
// [0,2048)      bins (radix-select histogram)
// [2048,3072)   counts[1024] (int)
// [3072,4096)   scales[1024]
// [4096,5120)   shifts[1024]
// [5120,5122)   select state {prefix, k}
// 5122 v1, 5123 v2, 5124 med, 5125 dyn, 5126 s, 5127 t
// 5128 maskSum (uint)
// [5136,6160)   partial[1024] (L1 block sums)
// [6160, 6160+nCheck)            x_sub
// [6160+nCheck, 6160+2*nCheck)   y_sub
// total ~= (6160 + 100000)*4 B ~= 425 KB of d_ws

__global__ void init_ws_kernel(unsigned* ws, int nwords) {
  int i = blockIdx.x * blockDim.x + threadIdx.x;
  if (i < nwords) ws[i] = 0u;
}

__global__ void compute_y_kernel(const float* __restrict__ dren,
                                 float* __restrict__ y,
                                 unsigned* maskSum, int n) {
  int i = blockIdx.x * blockDim.x + threadIdx.x;
  int stride = gridDim.x * blockDim.x;
  unsigned local = 0u;
  for (; i < n; i += stride) {
    float d = dren[i];
    y[i] = 1.0f / (d + 1e-6f);
    bool m = (d > 0.1f) && (d < 100.0f) && isfinite(d);
    local += m ? 1u : 0u;
  }
  local += __shfl_xor(local, 16, 32);
  local += __shfl_xor(local, 8, 32);
  local += __shfl_xor(local, 4, 32);
  local += __shfl_xor(local, 2, 32);
  local += __shfl_xor(local, 1, 32);
  if ((threadIdx.x & 31) == 0 && local) atomicAdd(maskSum, local);
}

__global__ void set_state_kernel(unsigned* state, unsigned k) {
  state[0] = 0u;
  state[1] = k;
}

// mode 0: key = bits(y[i]); mode 1: key = bits(|y[i]-med|).
// Positive floats are order-isomorphic to their uint32 bit patterns.
__global__ void hist_kernel(const float* __restrict__ y, const float* medp,
                            int mode, unsigned* __restrict__ bins,
                            const unsigned* state, unsigned maskHi, int shift,
                            int nbins, int n) {
  __shared__ unsigned h[2048];
  for (int b = threadIdx.x; b < nbins; b += blockDim.x) h[b] = 0u;
  __syncthreads();
  unsigned prefix = state[0] & maskHi;
  float med = (mode == 1) ? *medp : 0.0f;
  int i = blockIdx.x * blockDim.x + threadIdx.x;
  int stride = gridDim.x * blockDim.x;
  for (; i < n; i += stride) {
    float v = y[i];
    if (mode == 1) v = fabsf(v - med);
    unsigned u = __float_as_uint(v);
    if ((u & maskHi) == prefix) atomicAdd(&h[(u >> shift) & (nbins - 1)], 1u);
  }
  __syncthreads();
  for (int b = threadIdx.x; b < nbins; b += blockDim.x) {
    unsigned c = h[b];
    if (c) atomicAdd(&bins[b], c);
  }
}

__global__ void scan_kernel(unsigned* bins, unsigned* state, int shift,
                            int nbins, float* outVal) {
  if (threadIdx.x == 0) {
    unsigned k = state[1];
    unsigned cum = 0;
    int bucket = nbins - 1;
    for (int b = 0; b < nbins; ++b) {
      unsigned c = bins[b];
      if (cum + c > k) { bucket = b; break; }
      cum += c;
    }
    state[1] = k - cum;
    state[0] |= ((unsigned)bucket) << shift;
    if (shift == 0 && outVal) *outVal = __uint_as_float(state[0]);
  }
  __syncthreads();
  for (int b = threadIdx.x; b < nbins; b += blockDim.x) bins[b] = 0u;
}

// applyThresh=0: out = 0.5*(a+b)  (median of even-N array)
// applyThresh=1: out = dyn = MAD*0.5, clamped to THRESH if < 1e-5
__global__ void combine2_kernel(const float* a, const float* b, float* out,
                                int applyThresh) {
  float v = 0.5f * (*a + *b);
  if (applyThresh) {
    v *= 0.5f;
    if (v < 1e-5f) v = 0.01f;
  }
  *out = v;
}

__global__ void gather_kernel(const float* __restrict__ x,
                              const float* __restrict__ y,
                              const int* __restrict__ sample_idx,
                              const int* __restrict__ check_idx,
                              float* __restrict__ xs, float* __restrict__ ys,
                              float* __restrict__ scales,
                              float* __restrict__ shifts, int nCheck,
                              int nIter) {
  int i = blockIdx.x * blockDim.x + threadIdx.x;
  if (i < nCheck) {
    int c = check_idx[i];
    xs[i] = x[c];
    ys[i] = y[c];
  }
  int j = i - nCheck;
  if (j >= 0 && j < 1024) {
    if (j < nIter) {
      int i0 = sample_idx[2 * j], i1 = sample_idx[2 * j + 1];
      float x1 = x[i0], x2 = x[i1];
      float y1 = y[i0], y2 = y[i1];
      float s = (y2 - y1) / (x2 - x1 + 1e-8f);
      scales[j] = s;
      shifts[j] = y1 - s * x1;
    } else {
      scales[j] = -1.0f;  // padding rows: never valid (scale<=0)
      shifts[j] = 0.0f;
    }
  }
}

// One wave per (row-tile, column-span). res = A(16x4) x B(4x16):
//   A row i  = [s_i, t_i, 1, 0]
//   B col j  = [x_j, 1, -y_j, 0]^T
// => res_ij = s_i*x_j + t_i - y_j. Threshold+count per row via D layout.
__global__ __launch_bounds__(32) void ransac_wmma_kernel(
    const float* __restrict__ xs, const float* __restrict__ ys,
    const float* __restrict__ scales, const float* __restrict__ shifts,
    const float* dynp, int* __restrict__ counts, int nCols) {
  int lane = threadIdx.x;
  bool hi = lane >= 16;
  int l15 = lane & 15;
  int rowBase = blockIdx.x * 16;

  float dyn = *dynp;
  float sv = scales[rowBase + l15];
  float tv = shifts[rowBase + l15];
  v2f a;
  a.x = hi ? 1.0f : sv;  // K2 row of A is the constant 1 (for +t... actually t via K1)
  a.y = hi ? 0.0f : tv;  // lanes 0-15: {K0=s, K1=t}; lanes 16-31: {K2=1, K3=0}

  int nTiles = nCols >> 4;  // 50000/16 = 3125, exact
  int nSplit = gridDim.y;
  int per = (nTiles + nSplit - 1) / nSplit;
  int t0 = blockIdx.y * per;
  int t1 = t0 + per;
  if (t1 > nTiles) t1 = nTiles;

  int acc[8];
#pragma unroll
  for (int r = 0; r < 8; ++r) acc[r] = 0;

  v8f cz = {};
  for (int t = t0; t < t1; ++t) {
    int col = (t << 4) + l15;
    float xv = xs[col];
    float yv = ys[col];
    v2f b;
    b.x = hi ? -yv : xv;   // lanes 0-15: K0 row = x; lanes 16-31: K2 row = -y
    b.y = hi ? 0.0f : 1.0f;  // lanes 0-15: K1 row = 1; lanes 16-31: K3 row = 0
    v8f d = __builtin_amdgcn_wmma_f32_16x16x4_f32(false, a, false, b, (short)0,
                                                  cz, false, false);
#pragma unroll
    for (int r = 0; r < 8; ++r) acc[r] += (fabsf(d[r]) < dyn) ? 1 : 0;
  }

  // Reduce within each 16-lane half (D VGPR r: lanes0-15 = row r, 16-31 = row r+8)
#pragma unroll
  for (int r = 0; r < 8; ++r) {
    int v = acc[r];
    v += __shfl_xor(v, 1, 32);
    v += __shfl_xor(v, 2, 32);
    v += __shfl_xor(v, 4, 32);
    v += __shfl_xor(v, 8, 32);
    acc[r] = v;
  }
  if (lane == 0) {
#pragma unroll
    for (int r = 0; r < 8; ++r) atomicAdd(&counts[rowBase + r], acc[r]);
  } else if (lane == 16) {
#pragma unroll
    for (int r = 0; r < 8; ++r) atomicAdd(&counts[rowBase + 8 + r], acc[r]);
  }
}

// First-max argmax via key = (count<<32) | (~index); valid iff scale>0.
__global__ void argmax_kernel(const int* __restrict__ counts,
                              const float* __restrict__ scales,
                              const float* __restrict__ shifts, float* sOut,
                              float* tOut) {
  __shared__ unsigned long long sk[1024];
  int i = threadIdx.x;
  unsigned long long key = 0ull;
  if (scales[i] > 0.0f) {
    unsigned c = (unsigned)counts[i];
    key = (((unsigned long long)c) << 32) |
          (unsigned long long)(0xFFFFFFFFu - (unsigned)i);
  }
  sk[i] = key;
  __syncthreads();
  for (int off = 512; off > 0; off >>= 1) {
    if (i < off) {
      unsigned long long o = sk[i + off];
      if (o > sk[i]) sk[i] = o;
    }
    __syncthreads();
  }
  if (i == 0) {
    unsigned long long best = sk[0];
    float s = 1.0f, t = 0.0f;
    if (best != 0ull) {
      unsigned idx = 0xFFFFFFFFu - (unsigned)(best & 0xFFFFFFFFull);
      s = scales[idx];
      t = shifts[idx];
    }
    *sOut = s;
    *tOut = t;
  }
}

__global__ void final_kernel(const float* __restrict__ dren,
                             const float* __restrict__ dpri, const float* sp,
                             const float* tp, const float* __restrict__ y,
                             float* __restrict__ pmd,
                             float* __restrict__ partial, int n) {
  __shared__ float red[256];
  float s = *sp, t = *tp;
  float local = 0.0f;
  int i = blockIdx.x * blockDim.x + threadIdx.x;
  int stride = gridDim.x * blockDim.x;
  for (; i < n; i += stride) {
    float d = dren[i];
    float al = s * dpri[i] + t;
    float yv = y[i];
    bool m = (d > 0.1f) && (d < 100.0f) && isfinite(d);
    local += m ? fabsf(al - yv) : 0.0f;
    float alc = (al < 1e-4f) ? 1e-4f : al;
    pmd[i] = 1.0f / alc;
  }
  red[threadIdx.x] = local;
  __syncthreads();
  for (int off = 128; off > 0; off >>= 1) {
    if (threadIdx.x < off) red[threadIdx.x] += red[threadIdx.x + off];
    __syncthreads();
  }
  if (threadIdx.x == 0) partial[blockIdx.x] = red[0];
}

__global__ void reduce_kernel(const float* __restrict__ partial,
                              const unsigned* maskSum, float* out0, int nb) {
  __shared__ float red[256];
  float v = 0.0f;
  for (int i = threadIdx.x; i < nb; i += 256) v += partial[i];
  red[threadIdx.x] = v;
  __syncthreads();
  for (int off = 128; off > 0; off >>= 1) {
    if (threadIdx.x < off) red[threadIdx.x] += red[threadIdx.x + off];
    __syncthreads();
  }
  if (threadIdx.x == 0) {
    float ms = (float)(*maskSum);
    if (ms < 1.0f) ms = 1.0f;
    *out0 = 0.5f * red[0] / ms;  // LAMBDA_L1 * l1_loss
  }
}

extern "C" void kernel_launch(void* const* d_in, const int* in_sizes, int n_in,
                              void* d_out, int out_size, void* d_ws,
                              size_t ws_size, hipStream_t stream) {
  (void)n_in; (void)out_size; (void)ws_size;
  const float* dren = (const float*)d_in[0];
  const float* dpri = (const float*)d_in[1];
  const int* sample_idx = (const int*)d_in[2];
  const int* check_idx = (const int*)d_in[3];
  int n = in_sizes[0];            // B*H*W = 2,073,600
  int nIter = in_sizes[2] / 2;    // 1000
  int nCheck = in_sizes[3];       // 50000 (multiple of 16)

  float* out = (float*)d_out;
  float* y = out + 1;          // target_inv_ren
  float* pmd = out + 1 + n;    // prior_metric_depth

  unsigned* W = (unsigned*)d_ws;
  unsigned* bins = W;
  int* counts = (int*)(W + 2048);
  float* scales = (float*)(W + 3072);
  float* shifts = (float*)(W + 4096);
  unsigned* state = W + 5120;
  float* v1 = (float*)(W + 5122);
  float* v2 = (float*)(W + 5123);
  float* med = (float*)(W + 5124);
  float* dyn = (float*)(W + 5125);
  float* sP = (float*)(W + 5126);
  float* tP = (float*)(W + 5127);
  unsigned* maskSum = W + 5128;
  float* partial = (float*)(W + 5136);
  float* xs = (float*)(W + 6160);
  float* ys = xs + nCheck;

  const int initWords = 6160;
  init_ws_kernel<<<(initWords + 255) / 256, 256, 0, stream>>>(W, initWords);
  compute_y_kernel<<<512, 256, 0, stream>>>(dren, y, maskSum, n);

  // exact radix-select of the k-th smallest (positive floats as uint32 keys)
  auto select = [&](int mode, unsigned k, float* dst) {
    set_state_kernel<<<1, 1, 0, stream>>>(state, k);
    hist_kernel<<<256, 256, 0, stream>>>(y, med, mode, bins, state, 0x00000000u, 21, 2048, n);
    scan_kernel<<<1, 256, 0, stream>>>(bins, state, 21, 2048, nullptr);
    hist_kernel<<<256, 256, 0, stream>>>(y, med, mode, bins, state, 0xFFE00000u, 10, 2048, n);
    scan_kernel<<<1, 256, 0, stream>>>(bins, state, 10, 2048, nullptr);
    hist_kernel<<<256, 256, 0, stream>>>(y, med, mode, bins, state, 0xFFFFFC00u, 0, 1024, n);
    scan_kernel<<<1, 256, 0, stream>>>(bins, state, 0, 1024, dst);
  };

  unsigned k1 = (unsigned)(n / 2 - 1), k2 = (unsigned)(n / 2);
  // median(y) = mean of the two middle order statistics (N even)
  select(0, k1, v1);
  select(0, k2, v2);
  combine2_kernel<<<1, 1, 0, stream>>>(v1, v2, med, 0);
  // dyn = median(|y - med|) * 0.5, thresholded
  select(1, k1, v1);
  select(1, k2, v2);
  combine2_kernel<<<1, 1, 0, stream>>>(v1, v2, dyn, 1);

  int gatherTot = nCheck + 1024;
  gather_kernel<<<(gatherTot + 255) / 256, 256, 0, stream>>>(
      dpri, y, sample_idx, check_idx, xs, ys, scales, shifts, nCheck, nIter);

  dim3 rg(64, 64);  // 64 row tiles (1024 padded rows) x 64 column spans
  ransac_wmma_kernel<<<rg, 32, 0, stream>>>(xs, ys, scales, shifts, dyn, counts, nCheck);

  argmax_kernel<<<1, 1024, 0, stream>>>(counts, scales, shifts, sP, tP);

  final_kernel<<<1024, 256, 0, stream>>>(dren, dpri, sP, tP, y, pmd, partial, n);
  reduce_kernel<<<1, 256, 0, stream>>>(partial, maskSum, out, 1024);
}